// PitchLoss_29291676958982
// MI455X (gfx1250) — compile-verified
//
#include <hip/hip_runtime.h>
#include <hip/hip_bf16.h>

// CDNA5 / gfx1250, wave32 only.
typedef float v2f __attribute__((ext_vector_type(2)));
typedef float v8f __attribute__((ext_vector_type(8)));

#define PL_THRESH 0.5f

__device__ __forceinline__ float pl_readlane(float v, int srcLane) {
    return __uint_as_float(__builtin_amdgcn_readlane(__float_as_uint(v), srcLane));
}

// One wave32 workgroup per 16 notes. Exact-f32 segmented reduction mapped onto
// V_WMMA_F32_16X16X4_F32: D(16x16) += A(16x4 mask) x B(4x16 [gen|ref|ones]).
// EXEC must stay all-ones across every WMMA: loop bounds are forced into SGPRs
// via readfirstlane (scalar loop control) and all lane-dependent choices are
// branchless (v_cndmask / v_fma only).
__global__ __launch_bounds__(32)
void pitchloss_notes_wmma(const float* __restrict__ gen,
                          const float* __restrict__ ref,
                          const int*   __restrict__ onset,
                          const int*   __restrict__ offset,
                          float*       __restrict__ block_counts,
                          int T, int N)
{
    const int lane  = threadIdx.x;        // 0..31
    const int m     = lane & 15;          // A: note row M; B: column N
    const int khalf = (lane >> 4) << 1;   // K sub-block: 0 (lanes 0-15), 2 (lanes 16-31)
    const int Tm1   = T - 1;

    // Per-lane note interval (lanes 16-31 duplicate lanes 0-15's notes).
    const int note  = blockIdx.x * 16 + m;
    const int safei = (note < N) ? note : (N - 1);
    int on  = onset[safei];
    int off = offset[safei];
    on  = (note < N) ? on  : 0x7FFFFFFF;  // padded note: mask never fires
    off = (note < N) ? off : 0;

    // Wave-uniform time window covering this block's 16 notes.
    int tmin = on, tmax = off;
#pragma unroll
    for (int o = 16; o > 0; o >>= 1) {
        const int a = __shfl_xor(tmin, o, 32);
        const int b = __shfl_xor(tmax, o, 32);
        tmin = (a < tmin) ? a : tmin;
        tmax = (b > tmax) ? b : tmax;
    }
    // Force into SGPRs so the trip loop is scalar-controlled (EXEC untouched).
    tmin = __builtin_amdgcn_readfirstlane(tmin);
    tmax = __builtin_amdgcn_readfirstlane(tmax);
    tmin = (tmin < 0) ? 0 : tmin;
    tmax = (tmax > T) ? T : tmax;

    // Branchless B-column selectors (loop-invariant 0/1 floats):
    const float selg = (m == 0) ? 1.0f : 0.0f;   // column 0: gen
    const float selr = (m == 1) ? 1.0f : 0.0f;   // column 1: ref
    const float sel1 = (m == 2) ? 1.0f : 0.0f;   // column 2: ones (msum)

    // Two independent accumulator chains (D adds linearly) -> back-to-back
    // WMMAs in flight instead of a serial C->C dependence every 4 frames.
    v8f c0 = {};
    v8f c1 = {};

    const int base0 = tmin & ~7;
    for (int base = base0; base < tmax; base += 8) {
#pragma unroll
        for (int half = 0; half < 2; ++half) {
            const int b4 = base + 4 * half;
            const int t0 = b4 + khalf;
            const int t1 = t0 + 1;

            // A fragment: temporal mask for this lane's note at frames t0, t1.
            const float a0 = (t0 >= on && t0 < off) ? 1.0f : 0.0f;
            const float a1 = (t1 >= on && t1 < off) ? 1.0f : 0.0f;

            // Unconditional clamped loads (mask zeroes out-of-window frames),
            // blended with FMAs so no lane-dependent control flow appears.
            const int tc0 = (t0 < Tm1) ? t0 : Tm1;
            const int tc1 = (t1 < Tm1) ? t1 : Tm1;
            const float g0 = gen[tc0], g1 = gen[tc1];
            const float r0 = ref[tc0], r1 = ref[tc1];
            const float b0 = fmaf(selg, g0, fmaf(selr, r0, sel1));
            const float b1 = fmaf(selg, g1, fmaf(selr, r1, sel1));

            const v2f av = {a0, a1};
            const v2f bv = {b0, b1};
            if (half == 0) {
                c0 = __builtin_amdgcn_wmma_f32_16x16x4_f32(
                        false, av, false, bv, (short)0, c0, false, false);
            } else {
                c1 = __builtin_amdgcn_wmma_f32_16x16x4_f32(
                        false, av, false, bv, (short)0, c1, false, false);
            }
        }
    }

    const v8f c = c0 + c1;   // exact: disjoint frame chunks

    // D layout: VGPR j holds M=j (lanes 0-15, N=lane) and M=j+8 (lanes 16-31).
    // Extract via v_readlane (uniform results, no LDS bpermute storm).
    float local = 0.0f;
#pragma unroll
    for (int j = 0; j < 8; ++j) {
        const float sg_lo = pl_readlane(c[j],  0);   // D[j][0]   sum mask*gen
        const float sr_lo = pl_readlane(c[j],  1);   // D[j][1]   sum mask*ref
        const float ct_lo = pl_readlane(c[j],  2);   // D[j][2]   msum
        const float sg_hi = pl_readlane(c[j], 16);   // D[j+8][0]
        const float sr_hi = pl_readlane(c[j], 17);   // D[j+8][1]
        const float ct_hi = pl_readlane(c[j], 18);   // D[j+8][2]
        // Means exactly as the reference: sum/msum each, then |difference|.
        const float d_lo = fabsf(sg_lo / ct_lo - sr_lo / ct_lo);
        const float d_hi = fabsf(sg_hi / ct_hi - sr_hi / ct_hi);
        // NaN (empty note) compares false, matching JAX semantics.
        local += (d_lo > PL_THRESH) ? 1.0f : 0.0f;
        local += (d_hi > PL_THRESH) ? 1.0f : 0.0f;
    }

    if (lane == 0) block_counts[blockIdx.x] = local;  // write, never accumulate
}

// Single-wave final reduction: loss = (sum of block counts) / N.
__global__ __launch_bounds__(32)
void pitchloss_reduce(const float* __restrict__ block_counts,
                      float* __restrict__ out, int nblocks, float invN)
{
    const int lane = threadIdx.x;
    float s = 0.0f;
    for (int i = lane; i < nblocks; i += 32) s += block_counts[i];
#pragma unroll
    for (int o = 16; o > 0; o >>= 1) s += __shfl_xor(s, o, 32);
    if (lane == 0) out[0] = s * invN;
}

extern "C" void kernel_launch(void* const* d_in, const int* in_sizes, int n_in,
                              void* d_out, int out_size, void* d_ws, size_t ws_size,
                              hipStream_t stream)
{
    const float* gen = (const float*)d_in[0];   // gen_f0 (T,) f32
    const float* ref = (const float*)d_in[1];   // t_f0   (T,) f32
    const int*   on  = (const int*)  d_in[2];   // onset_times  (N,) i32
    const int*   off = (const int*)  d_in[3];   // offset_times (N,) i32
    const int T = in_sizes[0];
    const int N = in_sizes[2];

    const int nblocks = (N + 15) / 16;          // 64 for N=1024
    float* partials = (float*)d_ws;             // nblocks floats, fully rewritten each call

    pitchloss_notes_wmma<<<nblocks, 32, 0, stream>>>(gen, ref, on, off, partials, T, N);
    pitchloss_reduce<<<1, 32, 0, stream>>>(partials, (float*)d_out, nblocks, 1.0f / (float)N);
}